// GATStudent_88338887344526
// MI455X (gfx1250) — compile-verified
//
#include <hip/hip_runtime.h>
#include <hip/hip_bf16.h>
#include <math.h>

#define NEG_SLOPE 0.2f
#define GAT_EPS 1e-16f

typedef float v2f __attribute__((ext_vector_type(2)));
typedef float v8f __attribute__((ext_vector_type(8)));

// ---------------------------------------------------------------------------
// WMMA GEMM: C[M,Nc] = A[M,K] * B[K,Nc], all row-major fp32.
// One wave computes a 16-row x 64-col strip (4 accumulators, A frag reused).
// Requires M%16==0, K%4==0, Nc%64==0 (holds: 50000=16*3125, K in {512,256},
// Nc in {256,64}).
// A 16x4 f32 frag: lane l holds M=l&15, K=(l>>4)*2 {+0,+1} in 2 VGPRs.
// B 4x16 f32 frag: lane l holds N=l&15, K=(l>>4)*2 {+0,+1} in 2 VGPRs.
// C/D 16x16 f32:   lane l, vgpr v holds M=v+(l>>4)*8, N=l&15.
// ---------------------------------------------------------------------------
__global__ void __launch_bounds__(128)
gat_gemm_wmma(const float* __restrict__ A, const float* __restrict__ B,
              float* __restrict__ C, int M, int K, int Nc, int totalWaves) {
  const int wave = (int)((blockIdx.x * blockDim.x + threadIdx.x) >> 5);
  const int lane = (int)(threadIdx.x & 31);
  if (wave >= totalWaves) return;            // whole wave exits together
  const int colGroups = Nc >> 6;
  const int rowTile = wave / colGroups;
  const int colBase = (wave - rowTile * colGroups) << 6;
  const int row  = (rowTile << 4) + (lane & 15);
  const int kSub = (lane >> 4) << 1;         // 0 or 2
  const int nLane = lane & 15;

  v8f acc0 = {}; v8f acc1 = {}; v8f acc2 = {}; v8f acc3 = {};
  const float* Arow = A + (long)row * K;
  for (int k0 = 0; k0 < K; k0 += 4) {
    v2f a;
    a.x = Arow[k0 + kSub];
    a.y = Arow[k0 + kSub + 1];
    const float* B0 = B + (long)(k0 + kSub) * Nc + colBase + nLane;
    const float* B1 = B0 + Nc;
    v2f b0 = { B0[0],  B1[0]  };
    v2f b1 = { B0[16], B1[16] };
    v2f b2 = { B0[32], B1[32] };
    v2f b3 = { B0[48], B1[48] };
    acc0 = __builtin_amdgcn_wmma_f32_16x16x4_f32(false, a, false, b0, (short)0, acc0, false, false);
    acc1 = __builtin_amdgcn_wmma_f32_16x16x4_f32(false, a, false, b1, (short)0, acc1, false, false);
    acc2 = __builtin_amdgcn_wmma_f32_16x16x4_f32(false, a, false, b2, (short)0, acc2, false, false);
    acc3 = __builtin_amdgcn_wmma_f32_16x16x4_f32(false, a, false, b3, (short)0, acc3, false, false);
  }
  const int mBase = (rowTile << 4) + ((lane >> 4) << 3);
  #pragma unroll
  for (int v = 0; v < 8; ++v) {
    float* Crow = C + (long)(mBase + v) * Nc + colBase + nLane;
    Crow[0]  = acc0[v];
    Crow[16] = acc1[v];
    Crow[32] = acc2[v];
    Crow[48] = acc3[v];
  }
}

// ---------------------------------------------------------------------------
// Per-node attention logits: asrc[n,h] = dot(h[n,h,:], a_src[h,:]) (same adst)
// ---------------------------------------------------------------------------
__global__ void gat_alpha(const float* __restrict__ h, const float* __restrict__ a_src,
                          const float* __restrict__ a_dst, float* __restrict__ asrc,
                          float* __restrict__ adst, int N, int H, int Cc) {
  int i = blockIdx.x * blockDim.x + threadIdx.x;      // i = n*H + hh
  if (i >= N * H) return;
  int hh = i % H;
  const float* hv = h + (long)i * Cc;
  const float* as = a_src + hh * Cc;
  const float* ad = a_dst + hh * Cc;
  float s = 0.f, d = 0.f;
  for (int c = 0; c < Cc; ++c) { float xv = hv[c]; s += xv * as[c]; d += xv * ad[c]; }
  asrc[i] = s;
  adst[i] = d;
}

// init emax=-inf, denom=0 over NH entries; zero agg over aggN entries
__global__ void gat_init(float* __restrict__ emax, float* __restrict__ denom, int NH,
                         float* __restrict__ agg, long aggN) {
  long i = (long)blockIdx.x * blockDim.x + threadIdx.x;
  if (i < NH) { emax[i] = -__builtin_huge_valf(); denom[i] = 0.f; }
  if (i < aggN) agg[i] = 0.f;
}

__device__ __forceinline__ void atomicMaxF(float* addr, float v) {
  if (v >= 0.f) atomicMax((int*)addr, __float_as_int(v));
  else          atomicMin((unsigned int*)addr, __float_as_uint(v));
}

__device__ __forceinline__ void edge_endpoints(const int* __restrict__ ei, int e, int E,
                                               int& s, int& d) {
  if (e < E) { s = ei[e]; d = ei[E + e]; } else { s = e - E; d = s; }
}

// Pass 1: segment max of LeakyReLU(asrc[src]+adst[dst]) over incoming edges
__global__ void gat_edge_max(const int* __restrict__ ei, int E, int N, int H,
                             const float* __restrict__ asrc, const float* __restrict__ adst,
                             float* __restrict__ emax) {
  int idx = blockIdx.x * blockDim.x + threadIdx.x;
  int Etot = E + N;
  if (idx >= Etot * H) return;
  int e = idx / H, h = idx - (idx / H) * H;
  int s, d; edge_endpoints(ei, e, E, s, d);
  float ev = asrc[s * H + h] + adst[d * H + h];
  ev = ev > 0.f ? ev : NEG_SLOPE * ev;
  atomicMaxF(&emax[d * H + h], ev);
}

// Pass 2: segment sum of exp(e - emax[dst])
__global__ void gat_edge_sum(const int* __restrict__ ei, int E, int N, int H,
                             const float* __restrict__ asrc, const float* __restrict__ adst,
                             const float* __restrict__ emax, float* __restrict__ denom) {
  int idx = blockIdx.x * blockDim.x + threadIdx.x;
  int Etot = E + N;
  if (idx >= Etot * H) return;
  int e = idx / H, h = idx - (idx / H) * H;
  int s, d; edge_endpoints(ei, e, E, s, d);
  float ev = asrc[s * H + h] + adst[d * H + h];
  ev = ev > 0.f ? ev : NEG_SLOPE * ev;
  atomicAdd(&denom[d * H + h], __expf(ev - emax[d * H + h]));
}

// Pass 3: wave per edge; 32 lanes cover width=H*Cc contiguous floats each step.
// alpha recomputed (cheap) instead of stored (saves 27MB + a round trip).
__global__ void __launch_bounds__(256)
gat_edge_agg(const int* __restrict__ ei, int E, int N, int H, int Cc,
             const float* __restrict__ h, const float* __restrict__ asrc,
             const float* __restrict__ adst, const float* __restrict__ emax,
             const float* __restrict__ denom, float* __restrict__ agg) {
  int wave = (int)((blockIdx.x * blockDim.x + threadIdx.x) >> 5);
  int lane = (int)(threadIdx.x & 31);
  int Etot = E + N;
  if (wave >= Etot) return;
  int s, d; edge_endpoints(ei, wave, E, s, d);
  int width = H * Cc;
  int per = width >> 5;
  const float* hs = h + (long)s * width;
  float* ad = agg + (long)d * width;
  for (int j = 0; j < per; ++j) {
    int v = lane + (j << 5);
    int hh = v / Cc;
    float ev = asrc[s * H + hh] + adst[d * H + hh];
    ev = ev > 0.f ? ev : NEG_SLOPE * ev;
    float alpha = __expf(ev - emax[d * H + hh]) / (denom[d * H + hh] + GAT_EPS);
    atomicAdd(&ad[v], alpha * hs[v]);
  }
}

// io = ELU(io + b[i % width]) in place
__global__ void gat_bias_elu(float* __restrict__ io, const float* __restrict__ b,
                             long total, int width) {
  long i = (long)blockIdx.x * blockDim.x + threadIdx.x;
  if (i >= total) return;
  float xv = io[i] + b[(int)(i % width)];
  io[i] = xv > 0.f ? xv : (__expf(xv) - 1.f);
}

// out = in + b[i % width]
__global__ void gat_bias_out(const float* __restrict__ in, const float* __restrict__ b,
                             float* __restrict__ out, long total, int width) {
  long i = (long)blockIdx.x * blockDim.x + threadIdx.x;
  if (i >= total) return;
  out[i] = in[i] + b[(int)(i % width)];
}

extern "C" void kernel_launch(void* const* d_in, const int* in_sizes, int n_in,
                              void* d_out, int out_size, void* d_ws, size_t ws_size,
                              hipStream_t stream) {
  const float* x   = (const float*)d_in[0];
  const int*   ei  = (const int*)d_in[1];
  const float* W1  = (const float*)d_in[2];
  const float* as1 = (const float*)d_in[3];
  const float* ad1 = (const float*)d_in[4];
  const float* b1  = (const float*)d_in[5];
  const float* W2  = (const float*)d_in[6];
  const float* as2 = (const float*)d_in[7];
  const float* ad2 = (const float*)d_in[8];
  const float* b2  = (const float*)d_in[9];
  float* out = (float*)d_out;

  const int IN_C = 512, HID = 256, HEADS = 8, HEAD_C = 32, OUT_C = 64;
  const int N = in_sizes[0] / IN_C;
  const int E = in_sizes[1] / 2;
  const int Etot = E + N;

  // Workspace layout (floats): h1[N*HID] | agg1[N*HID] | asrc,adst,emax,denom[N*HEADS each]
  // h1 region is reused after layer 1 for h2[N*OUT_C] and agg2[N*OUT_C].
  float* ws    = (float*)d_ws;
  float* h1    = ws;
  float* agg1  = ws + (size_t)N * HID;
  float* sc    = agg1 + (size_t)N * HID;
  float* asrc  = sc;
  float* adst  = sc + (size_t)N * HEADS;
  float* emax  = sc + 2 * (size_t)N * HEADS;
  float* denom = sc + 3 * (size_t)N * HEADS;
  float* h2    = h1;                           // h1 dead after agg1 built
  float* agg2  = h1 + (size_t)N * OUT_C;

  // ---- Layer 1 ----
  {
    int waves = (N / 16) * (HID / 64);
    gat_gemm_wmma<<<(waves + 3) / 4, 128, 0, stream>>>(x, W1, h1, N, IN_C, HID, waves);
  }
  gat_alpha<<<(N * HEADS + 255) / 256, 256, 0, stream>>>(h1, as1, ad1, asrc, adst, N, HEADS, HEAD_C);
  {
    long aggN = (long)N * HID;
    gat_init<<<(int)((aggN + 255) / 256), 256, 0, stream>>>(emax, denom, N * HEADS, agg1, aggN);
  }
  gat_edge_max<<<(Etot * HEADS + 255) / 256, 256, 0, stream>>>(ei, E, N, HEADS, asrc, adst, emax);
  gat_edge_sum<<<(Etot * HEADS + 255) / 256, 256, 0, stream>>>(ei, E, N, HEADS, asrc, adst, emax, denom);
  gat_edge_agg<<<(Etot + 7) / 8, 256, 0, stream>>>(ei, E, N, HEADS, HEAD_C, h1, asrc, adst, emax, denom, agg1);
  gat_bias_elu<<<(int)(((long)N * HID + 255) / 256), 256, 0, stream>>>(agg1, b1, (long)N * HID, HID);

  // ---- Layer 2 ----
  {
    int waves = (N / 16) * (OUT_C / 64);
    gat_gemm_wmma<<<(waves + 3) / 4, 128, 0, stream>>>(agg1, W2, h2, N, HID, OUT_C, waves);
  }
  gat_alpha<<<(N + 255) / 256, 256, 0, stream>>>(h2, as2, ad2, asrc, adst, N, 1, OUT_C);
  {
    long aggN = (long)N * OUT_C;
    gat_init<<<(int)((aggN + 255) / 256), 256, 0, stream>>>(emax, denom, N, agg2, aggN);
  }
  gat_edge_max<<<(Etot + 255) / 256, 256, 0, stream>>>(ei, E, N, 1, asrc, adst, emax);
  gat_edge_sum<<<(Etot + 255) / 256, 256, 0, stream>>>(ei, E, N, 1, asrc, adst, emax, denom);
  gat_edge_agg<<<(Etot + 7) / 8, 256, 0, stream>>>(ei, E, N, 1, OUT_C, h2, asrc, adst, emax, denom, agg2);
  gat_bias_out<<<(int)(((long)N * OUT_C + 255) / 256), 256, 0, stream>>>(agg2, b2, out, (long)N * OUT_C, OUT_C);
}